// HAN_metapath_specific_20770461843681
// MI455X (gfx1250) — compile-verified
//
#include <hip/hip_runtime.h>
#include <math.h>

#define N_DIS 20000
#define N_MIC 80000
#define NTOT  (N_DIS + N_MIC)
#define NEDGE 1600000
#define FATT  64
#define SLOPE 0.2f

typedef float v2f __attribute__((ext_vector_type(2)));
typedef float v8f __attribute__((ext_vector_type(8)));

// ---------------------------------------------------------------------------
// GEMM: Z[m, 0:64] = X[m, 0:K] @ W[K, 0:64], fp32 via V_WMMA_F32_16X16X4_F32.
//
// Block = 128 threads = 4 waves; each wave owns a 32x64 output tile
// (two 16-row M-tiles x four 16-col N-tiles = 8 v8f accumulators).
// Both X (128x64) and W (64x64) k-slices are staged through LDS, ZERO-PADDED
// past M and K, so the WMMA inner loop is branch-free (pure ds_load + wmma)
// and runs a constant 16 k-steps per slice (padded zeros contribute nothing).
//
// fp32 WMMA fragment layouts (ISA 7.12.2):
//   A 16x4  : lanes 0-15 row M=lane, v0=K+0, v1=K+1 ; lanes 16-31 v0=K+2, v1=K+3
//   B 4x16  : v0 = rows {K+0 (lanes 0-15), K+2 (lanes 16-31)}, v1 = {K+1, K+3}
//   C 16x16 : VGPR r: lanes 0-15 -> M=r, N=lane ; lanes 16-31 -> M=r+8
// ---------------------------------------------------------------------------
__global__ __launch_bounds__(128) void gemm_wmma_f32(
        const float* __restrict__ X, const float* __restrict__ W,
        float* __restrict__ Z, int M, int K) {
    const int KT = 64;                       // k-slice per stage
    const int MT = 128;                      // rows per block
    __shared__ float lX[MT * KT];            // 32 KB, zero-padded
    __shared__ float lW[KT * FATT];          // 16 KB, zero-padded

    const int lane  = threadIdx.x & 31;
    const int wave  = threadIdx.x >> 5;
    const int row   = lane & 15;             // A row within 16-tile / C col
    const int khalf = lane >> 4;             // 0: K+0,1   1: K+2,3
    const int n     = lane & 15;

    const int blk_row = blockIdx.x * MT;     // first row of this block
    const int lr0 = wave * 32 + row;         // local A row, M-tile 0
    const int lr1 = lr0 + 16;                // local A row, M-tile 1

    v8f c0 = {}, c1 = {}, c2 = {}, c3 = {};  // M-tile 0, N-tiles 0..3
    v8f c4 = {}, c5 = {}, c6 = {}, c7 = {};  // M-tile 1, N-tiles 0..3

    for (int kt = 0; kt < K; kt += KT) {
        // ---- stage W[kt:kt+64, 0:64] (zero pad past K) ----
        for (int i = threadIdx.x; i < KT * FATT; i += blockDim.x) {
            int kg = kt + (i >> 6);
            lW[i] = (kg < K) ? W[(size_t)kg * FATT + (i & 63)] : 0.0f;
        }
        // ---- stage X[blk_row:blk_row+128, kt:kt+64] (zero pad past M,K) ----
        for (int i = threadIdx.x; i < MT * KT; i += blockDim.x) {
            int r  = i >> 6;                 // 0..127
            int cc = i & 63;                 // 0..63
            int mg = blk_row + r;
            int kg = kt + cc;
            lX[i] = (mg < M && kg < K) ? X[(size_t)mg * K + kg] : 0.0f;
        }
        __syncthreads();

        #pragma unroll 4
        for (int k0 = 0; k0 < KT; k0 += 4) {
            const int ka = k0 + 2 * khalf;   // even -> 8B aligned LDS reads
            v2f a0 = *(const v2f*)(lX + lr0 * KT + ka);
            v2f a1 = *(const v2f*)(lX + lr1 * KT + ka);

            const float* wr0 = lW + ka * FATT + n;        // row ka
            const float* wr1 = lW + (ka + 1) * FATT + n;  // row ka+1
            v2f b0 = { wr0[0],  wr1[0]  };
            v2f b1 = { wr0[16], wr1[16] };
            v2f b2 = { wr0[32], wr1[32] };
            v2f b3 = { wr0[48], wr1[48] };

            c0 = __builtin_amdgcn_wmma_f32_16x16x4_f32(false, a0, false, b0, (short)0, c0, false, false);
            c1 = __builtin_amdgcn_wmma_f32_16x16x4_f32(false, a0, false, b1, (short)0, c1, false, false);
            c2 = __builtin_amdgcn_wmma_f32_16x16x4_f32(false, a0, false, b2, (short)0, c2, false, false);
            c3 = __builtin_amdgcn_wmma_f32_16x16x4_f32(false, a0, false, b3, (short)0, c3, false, false);
            c4 = __builtin_amdgcn_wmma_f32_16x16x4_f32(false, a1, false, b0, (short)0, c4, false, false);
            c5 = __builtin_amdgcn_wmma_f32_16x16x4_f32(false, a1, false, b1, (short)0, c5, false, false);
            c6 = __builtin_amdgcn_wmma_f32_16x16x4_f32(false, a1, false, b2, (short)0, c6, false, false);
            c7 = __builtin_amdgcn_wmma_f32_16x16x4_f32(false, a1, false, b3, (short)0, c7, false, false);
        }
        __syncthreads();
    }

    // ---- store: C VGPR r holds row (tile_base + r + 8*khalf) ----
    const int m0 = blk_row + wave * 32 + 8 * khalf;       // M-tile 0
    #pragma unroll
    for (int r = 0; r < 8; ++r) {
        int mm = m0 + r;
        if (mm < M) {
            float* zr = Z + (size_t)mm * FATT + n;
            zr[0]  = c0[r];
            zr[16] = c1[r];
            zr[32] = c2[r];
            zr[48] = c3[r];
        }
        int mm1 = mm + 16;                                // M-tile 1
        if (mm1 < M) {
            float* zr = Z + (size_t)mm1 * FATT + n;
            zr[0]  = c4[r];
            zr[16] = c5[r];
            zr[32] = c6[r];
            zr[48] = c7[r];
        }
    }
}

// ---------------------------------------------------------------------------
// Init: zero h (d_out), denom; m_ord=0 (monotone encoding of "most negative").
// ---------------------------------------------------------------------------
__global__ void init_state(unsigned int* __restrict__ m_ord, float* __restrict__ denom,
                           float* __restrict__ h) {
    long long i = (long long)blockIdx.x * blockDim.x + threadIdx.x;
    if (i < (long long)NTOT * FATT) h[i] = 0.0f;
    if (i < NTOT) { m_ord[i] = 0u; denom[i] = 0.0f; }
}

__device__ __forceinline__ unsigned int float_to_ord(float f) {
    unsigned int u = __float_as_uint(f);
    return (u & 0x80000000u) ? ~u : (u | 0x80000000u);
}
__device__ __forceinline__ float ord_to_float(unsigned int u) {
    return __uint_as_float((u & 0x80000000u) ? (u ^ 0x80000000u) : ~u);
}

// ---------------------------------------------------------------------------
// Edge score: wave per edge, float2 per lane; e = leaky_relu(dot(zs, zd));
// atomicMax of monotone-encoded e into m_ord[dst].
// ---------------------------------------------------------------------------
__global__ void edge_score(const float* __restrict__ z, const int* __restrict__ src,
                           const int* __restrict__ dst, float* __restrict__ e,
                           unsigned int* __restrict__ m_ord) {
    const int lane = threadIdx.x & 31;
    const int wave = threadIdx.x >> 5;
    long long edge = (long long)blockIdx.x * (blockDim.x >> 5) + wave;
    if (edge >= NEDGE) return;
    const int s = src[edge], d = dst[edge];
    const float2* zs = (const float2*)(z + (size_t)s * FATT);
    const float2* zd = (const float2*)(z + (size_t)d * FATT);
    float2 a = zs[lane], b = zd[lane];
    float v = a.x * b.x + a.y * b.y;
    #pragma unroll
    for (int off = 16; off > 0; off >>= 1) v += __shfl_xor(v, off, 32);
    if (lane == 0) {
        float ev = v > 0.0f ? v : SLOPE * v;
        e[edge] = ev;
        atomicMax(&m_ord[d], float_to_ord(ev));
    }
}

// ---------------------------------------------------------------------------
// e_exp = exp(e - m[dst]) stored in place; denom[dst] += e_exp.
// ---------------------------------------------------------------------------
__global__ void edge_expsum(const int* __restrict__ dst, float* __restrict__ e,
                            const unsigned int* __restrict__ m_ord,
                            float* __restrict__ denom) {
    long long i = (long long)blockIdx.x * blockDim.x + threadIdx.x;
    if (i >= NEDGE) return;
    const int d = dst[i];
    const float m = ord_to_float(m_ord[d]);
    const float ex = __expf(e[i] - m);
    e[i] = ex;
    atomicAdd(&denom[d], ex);
}

// ---------------------------------------------------------------------------
// h[dst] += (e_exp/denom[dst]) * zs  — wave per edge, 2 features per lane.
// ---------------------------------------------------------------------------
__global__ void edge_scatter(const float* __restrict__ z, const int* __restrict__ src,
                             const int* __restrict__ dst, const float* __restrict__ e_exp,
                             const float* __restrict__ denom, float* __restrict__ h) {
    const int lane = threadIdx.x & 31;
    const int wave = threadIdx.x >> 5;
    long long edge = (long long)blockIdx.x * (blockDim.x >> 5) + wave;
    if (edge >= NEDGE) return;
    const int s = src[edge], d = dst[edge];
    const float alpha = e_exp[edge] / denom[d];
    const float2* zs = (const float2*)(z + (size_t)s * FATT);
    float2 a = zs[lane];
    float* hd = h + (size_t)d * FATT + 2 * lane;
    atomicAdd(hd + 0, alpha * a.x);
    atomicAdd(hd + 1, alpha * a.y);
}

// ---------------------------------------------------------------------------
// ELU in place on d_out.
// ---------------------------------------------------------------------------
__global__ void elu_inplace(float* __restrict__ h) {
    long long i = (long long)blockIdx.x * blockDim.x + threadIdx.x;
    if (i >= (long long)NTOT * FATT) return;
    const float v = h[i];
    h[i] = v > 0.0f ? v : (__expf(v) - 1.0f);
}

extern "C" void kernel_launch(void* const* d_in, const int* in_sizes, int n_in,
                              void* d_out, int out_size, void* d_ws, size_t ws_size,
                              hipStream_t stream) {
    const float* d_sim  = (const float*)d_in[0];   // [20000, 383]
    const float* mi_sim = (const float*)d_in[1];   // [80000, 495]
    const float* W_d    = (const float*)d_in[2];   // [383, 64]
    const float* W_mi   = (const float*)d_in[3];   // [495, 64]
    const int*   src    = (const int*)d_in[4];     // [E]
    const int*   dst    = (const int*)d_in[5];     // [E]
    float*       out    = (float*)d_out;           // [N, 64]

    // Workspace layout
    float* z            = (float*)d_ws;                         // NTOT*64
    float* e            = z + (size_t)NTOT * FATT;              // NEDGE (scores -> e_exp)
    unsigned int* m_ord = (unsigned int*)(e + NEDGE);           // NTOT
    float* denom        = (float*)(m_ord + NTOT);               // NTOT

    const long long tot_h = (long long)NTOT * FATT;

    // 0) init accumulators
    init_state<<<(int)((tot_h + 255) / 256), 256, 0, stream>>>(m_ord, denom, out);

    // 1) projections via fp32 WMMA (4 waves/block, 128 rows/block)
    gemm_wmma_f32<<<(N_DIS + 127) / 128, 128, 0, stream>>>(d_sim, W_d, z, N_DIS, 383);
    gemm_wmma_f32<<<(N_MIC + 127) / 128, 128, 0, stream>>>(mi_sim, W_mi,
                                                           z + (size_t)N_DIS * FATT, N_MIC, 495);

    // 2) edge scores + segment max (wave per edge)
    edge_score<<<(NEDGE + 7) / 8, 256, 0, stream>>>(z, src, dst, e, m_ord);

    // 3) exp + segment sum (thread per edge)
    edge_expsum<<<(NEDGE + 255) / 256, 256, 0, stream>>>(dst, e, m_ord, denom);

    // 4) normalized weighted scatter (wave per edge)
    edge_scatter<<<(NEDGE + 7) / 8, 256, 0, stream>>>(z, src, dst, e, denom, out);

    // 5) ELU
    elu_inplace<<<(int)((tot_h + 255) / 256), 256, 0, stream>>>(out);
}